// ACELoss_19378892439658
// MI455X (gfx1250) — compile-verified
//
#include <hip/hip_runtime.h>
#include <hip/hip_bf16.h>
#include <math.h>

#define EPS    1e-5f
#define C_NUM  128
#define T_NUM  64
#define CT     8192          // C_NUM * T_NUM floats per sample (32 KB)
#define BLK    256
#define GRID_MAIN 1024

// Low 32 bits of a generic pointer to LDS == LDS byte offset.
__device__ __forceinline__ uint32_t lds_addr_of(const void* p) {
  return (uint32_t)(uintptr_t)p;
}

// Stream one 32 KB sample tile (8192 floats) global -> LDS with 8 async b128 per thread.
// GV mode: global_load_async_to_lds_b128 vdst(LDS byte addr), v[addr:addr+1], off
// th:TH_LOAD_NT — x is a 268 MB read-once stream (> 192 MB L2): don't thrash the cache.
__device__ __forceinline__ void async_load_tile(uint32_t lds_base, const float* g, int tid) {
  uint32_t l = lds_base + (uint32_t)tid * 16u;
  const float* gp = g + tid * 4;
#pragma unroll
  for (int k = 0; k < 8; ++k) {
    asm volatile("global_load_async_to_lds_b128 %0, %1, off th:TH_LOAD_NT"
                 :: "v"(l + (uint32_t)(k * 4096)), "v"(gp + k * 1024)
                 : "memory");
  }
}

// ---------------- exclusive prefix sum of target_lengths -> starts ----------------
__global__ void ace_scan(const int* __restrict__ lens, int* __restrict__ starts, int B) {
  __shared__ int s[BLK];
  const int tid = threadIdx.x;
  const int chunk = (B + BLK - 1) / BLK;
  const int base = tid * chunk;
  int sum = 0;
  for (int i = 0; i < chunk; ++i) {
    int idx = base + i;
    if (idx < B) sum += lens[idx];
  }
  s[tid] = sum;
  for (int off = 1; off < BLK; off <<= 1) {
    __syncthreads();
    int v = (tid >= off) ? s[tid - off] : 0;
    __syncthreads();
    s[tid] += v;
  }
  __syncthreads();
  int run = (tid == 0) ? 0 : s[tid - 1];
  for (int i = 0; i < chunk; ++i) {
    int idx = base + i;
    if (idx < B) { starts[idx] = run; run += lens[idx]; }
  }
}

// ---------------- main: per-sample argmax histogram + ACE loss ----------------
__global__ void __launch_bounds__(BLK) ace_main(const float* __restrict__ x,
                                                const int* __restrict__ y,
                                                const int* __restrict__ lens,
                                                const int* __restrict__ starts,
                                                float* __restrict__ losses, int B) {
  __shared__ float tile[2][CT];         // 2 x 32 KB double buffer
  __shared__ float pmax[T_NUM * 4];
  __shared__ int   pidx[T_NUM * 4];
  __shared__ int   hist_n[C_NUM];
  __shared__ int   hist_y[C_NUM];
  __shared__ float red[BLK];

  const int tid = threadIdx.x;
  int b = blockIdx.x;
  if (b >= B) return;                   // uniform per block

  const uint32_t tbase0 = lds_addr_of(&tile[0][0]);
  const uint32_t tbase1 = lds_addr_of(&tile[1][0]);
  int cur = 0;
  async_load_tile(tbase0, x + (size_t)b * CT, tid);

  for (; b < B; b += gridDim.x) {
    const int  nb      = b + gridDim.x;
    const bool hasnext = nb < B;
    if (hasnext)
      async_load_tile(cur ? tbase0 : tbase1, x + (size_t)nb * CT, tid);

    if (tid < C_NUM) { hist_n[tid] = 0; hist_y[tid] = 0; }

    // retire the current tile's 8 async loads (the 8 newest belong to next tile)
    if (hasnext) asm volatile("s_wait_asynccnt 0x8" ::: "memory");
    else         asm volatile("s_wait_asynccnt 0x0" ::: "memory");
    __syncthreads();

    // ---- argmax over C per timestep: thread = (t, c-group of 32) ----
    {
      const int t  = tid & 63;
      const int cg = tid >> 6;                         // 0..3
      const float* tp = &tile[cur][(cg * 32) * T_NUM + t];
      float best = tp[0];
      int   bi   = cg * 32;
#pragma unroll 4
      for (int i = 1; i < 32; ++i) {
        float v = tp[i * T_NUM];                       // stride-64 dwords: conflict-free
        if (v > best) { best = v; bi = cg * 32 + i; }  // strict > keeps lowest index
      }
      pmax[t * 4 + cg] = best;
      pidx[t * 4 + cg] = bi;
    }
    __syncthreads();

    const int len   = lens[b];
    const int start = starts[b];

    if (tid < T_NUM) {                                 // reduce 4 c-groups (ascending c)
      float m  = pmax[tid * 4];
      int   mi = pidx[tid * 4];
#pragma unroll
      for (int g = 1; g < 4; ++g) {
        float v = pmax[tid * 4 + g];
        if (v > m) { m = v; mi = pidx[tid * 4 + g]; }
      }
      atomicAdd(&hist_n[mi], 1);
    }
    for (int i = tid; i < len; i += BLK)               // target histogram
      atomicAdd(&hist_y[y[start + i]], 1);
    __syncthreads();

    // ---- masked loss over classes ----
    float yk = 0.f, nkm = 0.f;
    if (tid < C_NUM) {
      yk = (float)hist_y[tid];
      if (yk != 0.f) nkm = (float)hist_n[tid];         // n_k masked by (y_k != 0)
    }
    red[tid] = nkm;
    for (int s = BLK / 2; s > 0; s >>= 1) {
      __syncthreads();
      if (tid < s) red[tid] += red[tid + s];
    }
    __syncthreads();
    const float n_sum = red[0];
    __syncthreads();

    float term = 0.f;
    if (tid < C_NUM && yk != 0.f) {
      float n_p = (n_sum == 0.f) ? EPS : fmaxf(nkm / n_sum, EPS);
      float y_p = yk / (float)len;                     // y_sum == target length
      term = -n_p * logf(y_p);
    }
    red[tid] = term;
    for (int s = BLK / 2; s > 0; s >>= 1) {
      __syncthreads();
      if (tid < s) red[tid] += red[tid + s];
    }
    __syncthreads();
    if (tid == 0) losses[b] = red[0];
    __syncthreads();                                   // protect red/hists & tile reuse

    cur ^= 1;
  }
}

// ---------------- mean over B (deterministic tree) ----------------
__global__ void ace_finish(const float* __restrict__ losses, float* __restrict__ out, int B) {
  __shared__ float red[BLK];
  const int tid = threadIdx.x;
  float s = 0.f;
  for (int i = tid; i < B; i += BLK) s += losses[i];
  red[tid] = s;
  for (int w = BLK / 2; w > 0; w >>= 1) {
    __syncthreads();
    if (tid < w) red[tid] += red[tid + w];
  }
  __syncthreads();
  if (tid == 0) out[0] = red[0] / (float)B;
}

extern "C" void kernel_launch(void* const* d_in, const int* in_sizes, int n_in,
                              void* d_out, int out_size, void* d_ws, size_t ws_size,
                              hipStream_t stream) {
  const float* x    = (const float*)d_in[0];   // [B, 128, 64] fp32
  const int*   y    = (const int*)d_in[1];     // [B*L] int32
  const int*   lens = (const int*)d_in[2];     // [B] int32
  const int B = in_sizes[2];

  int*   starts = (int*)d_ws;
  float* losses = (float*)((char*)d_ws + (size_t)B * sizeof(int));

  ace_scan<<<1, BLK, 0, stream>>>(lens, starts, B);
  int grid = (B < GRID_MAIN) ? B : GRID_MAIN;
  ace_main<<<grid, BLK, 0, stream>>>(x, y, lens, starts, losses, B);
  ace_finish<<<1, BLK, 0, stream>>>(losses, (float*)d_out, B);
}